// DeTPPLoss_90735479095886
// MI455X (gfx1250) — compile-verified
//
#include <hip/hip_runtime.h>
#include <hip/hip_bf16.h>

typedef __attribute__((ext_vector_type(2))) float v2f;
typedef __attribute__((ext_vector_type(4))) float v4f;
typedef __attribute__((ext_vector_type(8))) float v8f;

#define LDIM 2048
#define BDIM 64
#define KDIM 4
#define CDIM 128
#define IDIM 512
#define NSAMP (IDIM * BDIM)   // 32768
#define WAVES 16
#define BLOCK (WAVES * 32)

// perms of {0,1,2,3} in lexicographic (itertools) order, packed 2 bits per
// position (perm[k] = (byte >> 2k) & 3), 8 perm-bytes per u64 immediate.
// bytes: 228,180,216,120,156,108,225,177 | 201,57,141,45,210,114,198,54 |
//        78,30,147,99,135,39,75,27
#define PERM_PACK0 0xB1E16C9C78D8B4E4ull
#define PERM_PACK1 0x36C672D22D8D39C9ull
#define PERM_PACK2 0x1B4B278763931E4Eull

__global__ __launch_bounds__(BLOCK)
void detpp_main(const float* __restrict__ in_time,
                const float* __restrict__ in_amount,
                const int*   __restrict__ in_mcc,
                const float* __restrict__ out_time,
                const float* __restrict__ out_amount,
                const float* __restrict__ out_logits,
                const float* __restrict__ presence,
                const int*   __restrict__ indices,
                const int*   __restrict__ subset_lengths,
                float* __restrict__ accum)
{
    __shared__ float cost_lds[16 * 17];   // 16 samples x 16 cost entries, stride 17
    __shared__ float mask_lds[16];
    __shared__ float leftover_lds[16];

    const int lane = threadIdx.x & 31;
    const int w    = threadIdx.x >> 5;          // wave id == sample-in-block
    const int n    = blockIdx.x * WAVES + w;    // global sample
    const int i    = n >> 6;                    // n / B
    const int b    = n & 63;                    // n % B
    const int idx  = indices[n];                // indices is (I,B) row-major
    const float maskf = (i < subset_lengths[b]) ? 1.0f : 0.0f;

    // Rolling window over inputs: positions (idx+t) mod L, t = 0..K (wave-uniform)
    float tw[KDIM + 1], aw[KDIM + 1];
    int   cw[KDIM + 1];
#pragma unroll
    for (int t = 0; t <= KDIM; ++t) {
        int pt = idx + t; if (pt >= LDIM) pt -= LDIM;
        tw[t] = in_time [pt * BDIM + b];
        aw[t] = in_amount[pt * BDIM + b];
        cw[t] = in_mcc  [pt * BDIM + b];
    }

    const int baseK = (idx * BDIM + b) * KDIM;          // 16B-aligned in floats
    const float4 ot4 = *reinterpret_cast<const float4*>(out_time   + baseK);
    const float4 oa4 = *reinterpret_cast<const float4*>(out_amount + baseK);
    const float4 ps4 = *reinterpret_cast<const float4*>(presence   + baseK);
    const float otk[4] = {ot4.x, ot4.y, ot4.z, ot4.w};
    const float oak[4] = {oa4.x, oa4.y, oa4.z, oa4.w};
    const float psk[4] = {ps4.x, ps4.y, ps4.z, ps4.w};

    const float* lbase = out_logits + (size_t)baseK * CDIM;

#pragma unroll
    for (int k = 0; k < KDIM; ++k) {
        // 32 lanes x 4 floats = 128 channels, coalesced global_load_b128.
        // Zero reuse across samples -> non-temporal to spare WGP$/L2.
        v4f v = __builtin_nontemporal_load(
            reinterpret_cast<const v4f*>(lbase + k * CDIM + lane * 4));

        float mx = fmaxf(fmaxf(v[0], v[1]), fmaxf(v[2], v[3]));
#pragma unroll
        for (int s = 16; s >= 1; s >>= 1) mx = fmaxf(mx, __shfl_xor(mx, s, 32));

        float se = expf(v[0] - mx) + expf(v[1] - mx) + expf(v[2] - mx) + expf(v[3] - mx);
#pragma unroll
        for (int s = 16; s >= 1; s >>= 1) se += __shfl_xor(se, s, 32);
        const float logZ = mx + logf(se);

        // pick logit of the true class for each target t (broadcast to all lanes)
        float pick[KDIM];
#pragma unroll
        for (int t = 0; t < KDIM; ++t) {
            const int c = cw[t + 1];
            const int e = c & 3;
            float sel = (e == 0) ? v[0] : (e == 1) ? v[1] : (e == 2) ? v[2] : v[3];
            pick[t] = __shfl(sel, c >> 2, 32);
        }

        if (lane < 4) {  // lane == t writes cost[k][t]
            const float pk = (lane == 0) ? pick[0] : (lane == 1) ? pick[1]
                           : (lane == 2) ? pick[2] : pick[3];
            const float tt = ((lane == 0) ? tw[1] : (lane == 1) ? tw[2]
                            : (lane == 2) ? tw[3] : tw[4]) - tw[0];
            const float at = (lane == 0) ? aw[1] : (lane == 1) ? aw[2]
                           : (lane == 2) ? aw[3] : aw[4];
            cost_lds[w * 17 + k * 4 + lane] =
                (logZ - pk) + fabsf(otk[k] - tt) + fabsf(oak[k] - at) - psk[k];
        }
    }

    if (lane == 0) {
        float sp = 0.f;   // -log_sigmoid(-x) == softplus(x), numerically stable
#pragma unroll
        for (int k = 0; k < KDIM; ++k)
            sp += fmaxf(psk[k], 0.f) + log1pf(expf(-fabsf(psk[k])));
        leftover_lds[w] = maskf * sp;
        mask_lds[w]     = maskf;
    }
    __syncthreads();

    // Wave 0: exact batched assignment for the block's 16 samples via WMMA.
    // perm_cost(16x24) = E(16x16) x Ind(16x24), split into 4 K=4 chunks and
    // 2 output tiles (perms 0..15 and perms 16..23, invalid cols seeded 1e30).
    if (w == 0) {
        const int half = lane >> 4;
        const int col  = lane & 15;
        const int row  = col;                // A-matrix M index for this lane

        // Branchless per-lane perm codes from packed immediates (pure VALU).
        const int shift = (col & 7) * 8;
        const int pm1 = (int)(((col < 8 ? PERM_PACK0 : PERM_PACK1) >> shift) & 0xFF);
        const int pm2 = (int)((PERM_PACK2 >> shift) & 0xFF);
        const bool has2 = (col < 8);

        v8f acc1 = {0.f, 0.f, 0.f, 0.f, 0.f, 0.f, 0.f, 0.f};
        v8f acc2 = {1e30f, 1e30f, 1e30f, 1e30f, 1e30f, 1e30f, 1e30f, 1e30f};

#pragma unroll
        for (int q = 0; q < 4; ++q) {
            v2f a;
            a.x = cost_lds[row * 17 + 4 * q + 2 * half];
            a.y = cost_lds[row * 17 + 4 * q + 2 * half + 1];

            const int s1 = (pm1 >> (2 * q)) & 3;
            const int s2 = (pm2 >> (2 * q)) & 3;
            v2f b1, b2;
            b1.x = (s1 == 2 * half)              ? 1.f : 0.f;
            b1.y = (s1 == 2 * half + 1)          ? 1.f : 0.f;
            b2.x = (has2 && s2 == 2 * half)      ? 1.f : 0.f;
            b2.y = (has2 && s2 == 2 * half + 1)  ? 1.f : 0.f;

            acc1 = __builtin_amdgcn_wmma_f32_16x16x4_f32(
                false, a, false, b1, (short)0, acc1, false, false);
            acc2 = __builtin_amdgcn_wmma_f32_16x16x4_f32(
                false, a, false, b2, (short)0, acc2, false, false);
        }

        // min over 24 permutations per sample; D vgpr v holds samples v / v+8
        float blockLoss = 0.f;
#pragma unroll
        for (int v = 0; v < 8; ++v) {
            float x = fminf(acc1[v], acc2[v]);
#pragma unroll
            for (int s = 8; s >= 1; s >>= 1) x = fminf(x, __shfl_xor(x, s, 32));
            const int samp = v + half * 8;
            if (col == 0) blockLoss += x * mask_lds[samp];
        }
        blockLoss += __shfl_xor(blockLoss, 16, 32);  // lane0 += lane16

        if (lane == 0) {
            float lv = blockLoss;
#pragma unroll
            for (int s2 = 0; s2 < 16; ++s2) lv += leftover_lds[s2];
            atomicAdd(accum, lv);
        }
    }
}

__global__ void detpp_final(const int* __restrict__ subset_lengths,
                            const float* __restrict__ accum,
                            float* __restrict__ out)
{
    if (threadIdx.x == 0 && blockIdx.x == 0) {
        float V = 0.f;
        for (int b = 0; b < BDIM; ++b) V += (float)subset_lengths[b];
        out[0] = accum[0] / V;
    }
}

extern "C" void kernel_launch(void* const* d_in, const int* in_sizes, int n_in,
                              void* d_out, int out_size, void* d_ws, size_t ws_size,
                              hipStream_t stream) {
    const float* in_time    = (const float*)d_in[0];
    const float* in_amount  = (const float*)d_in[1];
    const int*   in_mcc     = (const int*)  d_in[2];
    const float* out_time   = (const float*)d_in[3];
    const float* out_amount = (const float*)d_in[4];
    const float* out_logits = (const float*)d_in[5];
    const float* presence   = (const float*)d_in[6];
    // d_in[7] = lengths (unused by the loss core)
    const int*   indices    = (const int*)  d_in[8];
    const int*   subset_len = (const int*)  d_in[9];

    float* accum = (float*)d_ws;
    (void)hipMemsetAsync(accum, 0, sizeof(float), stream);

    detpp_main<<<NSAMP / WAVES, BLOCK, 0, stream>>>(
        in_time, in_amount, in_mcc, out_time, out_amount, out_logits,
        presence, indices, subset_len, accum);

    detpp_final<<<1, 64, 0, stream>>>(subset_len, accum, (float*)d_out);
}